// YOLOv1Loss_62569083568616
// MI455X (gfx1250) — compile-verified
//
#include <hip/hip_runtime.h>
#include <cstdint>

// YOLOv1 loss for MI455X (gfx1250).
// Memory-bound: 24 MB pred stream -> ~1.0 us at 23.3 TB/s HBM (L2-resident on
// graph replay, so effectively L2-bandwidth bound). Strategy:
//   - async global->LDS staging (CDNA5 ASYNCcnt path) for perfect coalescing
//     of the 120 B/cell records, one cell per thread,
//   - wave32 reduction on the matrix pipe (V_WMMA_F32_16X16X4_F32 with a
//     ones B-matrix => row sums),
//   - deterministic two-stage final reduction (no float atomics).

#define S_GRID   14
#define NBOX     2
#define NCLS     20
#define DCH      30                    // 5*NB + C
#define CPI      (S_GRID * S_GRID)     // 196 cells per image
#define BLOCK    256
#define TILE_F   (BLOCK * DCH)         // 7680 floats staged per block
#define TILE_B   (TILE_F * 4)          // 30720 bytes
#define CHUNKS   (TILE_B / 16)         // 1920 x b128 lane-transfers

typedef __attribute__((ext_vector_type(2))) float v2f;
typedef __attribute__((ext_vector_type(8))) float v8f;

// ---- wave32 reduction via the matrix pipe -----------------------------------
// A (16x4): lane L<16 holds {x[L], 0} in K0/K1; lane L>=16 holds {x[L], 0} in
// K2/K3 for row M=L-16.  B = ones(4x16).  D[m][n] = x[m] + x[m+16] for all n.
// Sum of the 8 D VGPRs in a lane = half-sum of rows; shfl_xor(16) completes it.
__device__ __forceinline__ float wave_reduce32(float x) {
#if __has_builtin(__builtin_amdgcn_wmma_f32_16x16x4_f32)
  v2f a; a[0] = x;    a[1] = 0.0f;
  v2f b; b[0] = 1.0f; b[1] = 1.0f;
  v8f c = {};
  v8f d = __builtin_amdgcn_wmma_f32_16x16x4_f32(
      /*neg_a=*/false, a, /*neg_b=*/false, b,
      /*c_mod=*/(short)0, c, /*reuse_a=*/false, /*reuse_b=*/false);
  float s = d[0] + d[1] + d[2] + d[3] + d[4] + d[5] + d[6] + d[7];
  s += __shfl_xor(s, 16, 32);
  return s;
#else
  #pragma unroll
  for (int m = 1; m < 32; m <<= 1) x += __shfl_xor(x, m, 32);
  return x;
#endif
}

// ---- kernel 1: zero the cell map --------------------------------------------
__global__ void yolo_zero(int* __restrict__ cellmap, int ncells) {
  int i = blockIdx.x * blockDim.x + threadIdx.x;
  if (i < ncells) cellmap[i] = 0;
}

// ---- kernel 2: scatter targets into the cell map (index+1, 0 = empty) -------
__global__ void yolo_encode(const float* __restrict__ targets, int ntgt,
                            int* __restrict__ cellmap, int ncells) {
  int t = blockIdx.x * blockDim.x + threadIdx.x;
  if (t >= ntgt) return;
  const float* r = targets + 6 * (size_t)t;
  int   b  = (int)r[0];
  float cx = (r[2] + r[4]) * 0.5f;
  float cy = (r[3] + r[5]) * 0.5f;
  int ci = (int)ceilf(cx * (float)S_GRID) - 1;   // matches jnp.ceil(cxy*S)-1
  int cj = (int)ceilf(cy * (float)S_GRID) - 1;
  ci = min(max(ci, 0), S_GRID - 1);
  cj = min(max(cj, 0), S_GRID - 1);
  int idx = b * CPI + ci * S_GRID + cj;          // cells unique per image
  if (idx >= 0 && idx < ncells) cellmap[idx] = t + 1;
}

// ---- kernel 3: main loss, one partial per block -----------------------------
__global__ __launch_bounds__(BLOCK)
void yolo_loss_main(const float* __restrict__ pred,
                    const float* __restrict__ targets,
                    const int*   __restrict__ cellmap,
                    float*       __restrict__ partials,
                    int ncells) {
  __shared__ float tile[TILE_F];
  __shared__ float wsum[BLOCK / 32];

  const int cell0 = blockIdx.x * BLOCK;

  // --- async stage pred tile into LDS (coalesced 16B/lane, ASYNCcnt path) ---
  {
    // SADDR = pred base (uniform SGPR pair); VADDR = 32-bit byte offset.
    uint64_t gbase    = (uint64_t)(uintptr_t)pred;
    uint32_t base_off = (uint32_t)cell0 * (uint32_t)(DCH * 4);
    // generic LDS pointer: high 32 bits = shared aperture, low 32 = LDS offset
    uint32_t lds_base = (uint32_t)(uintptr_t)(void*)tile;
    int nch = CHUNKS;
    if (ncells - cell0 < BLOCK)                  // defensive (never hit: 200704%256==0)
      nch = ((ncells - cell0) * DCH * 4) / 16;
    for (int i = threadIdx.x; i < nch; i += BLOCK) {
      uint32_t off = (uint32_t)i * 16u;
      uint32_t la  = lds_base + off;
      uint32_t ga  = base_off + off;
      asm volatile("global_load_async_to_lds_b128 %0, %1, %2"
                   :: "v"(la), "v"(ga), "s"(gbase) : "memory");
    }
    asm volatile("s_wait_asynccnt 0" ::: "memory");
  }
  __syncthreads();

  // --- per-cell loss ---
  const int cell = cell0 + threadIdx.x;
  float l = 0.0f;
  if (cell < ncells) {
    const float* p = tile + (size_t)threadIdx.x * DCH;
    int t = cellmap[cell];
    if (t > 0) {
      const float* r = targets + 6 * (size_t)(t - 1);
      float cx = (r[2] + r[4]) * 0.5f, cy = (r[3] + r[5]) * 0.5f;
      float tw = r[4] - r[2],          th = r[5] - r[3];
      float sx = cx * (float)S_GRID,   sy = cy * (float)S_GRID;
      float offx = sx - (ceilf(sx) - 1.0f);
      float offy = sy - (ceilf(sy) - 1.0f);
      // target corners (reference's _corners on the encoded box)
      float txc = offx / (float)S_GRID, tyc = offy / (float)S_GRID;
      float tminx = txc - tw * 0.5f, tmaxx = txc + tw * 0.5f;
      float tminy = tyc - th * 0.5f, tmaxy = tyc + th * 0.5f;
      float areat = (tmaxx - tminx) * (tmaxy - tminy);

      float ious[NBOX];
      #pragma unroll
      for (int k = 0; k < NBOX; ++k) {
        float bx = p[5*k], by = p[5*k+1], bw = p[5*k+2], bh = p[5*k+3];
        float pxc = bx / (float)S_GRID, pyc = by / (float)S_GRID;
        float pminx = pxc - bw * 0.5f, pmaxx = pxc + bw * 0.5f;
        float pminy = pyc - bh * 0.5f, pmaxy = pyc + bh * 0.5f;
        float iw = fmaxf(fminf(pmaxx, tmaxx) - fmaxf(pminx, tminx), 0.0f);
        float ih = fmaxf(fminf(pmaxy, tmaxy) - fmaxf(pminy, tminy), 0.0f);
        float inter = iw * ih;
        float areap = (pmaxx - pminx) * (pmaxy - pminy);
        ious[k] = inter / (areap + areat - inter);
      }
      int   sel  = (ious[0] >= ious[1]) ? 0 : 1;   // argmax, first max wins
      float miou = fmaxf(ious[0], ious[1]);
      const float* sp = p + 5 * sel;

      float dx = sp[0] - offx, dy = sp[1] - offy;
      float dw = sqrtf(sp[2]) - sqrtf(tw);
      float dh = sqrtf(sp[3]) - sqrtf(th);
      l = 5.0f * (dx*dx + dy*dy + dw*dw + dh*dh);   // L_COORD*(xy+wh)
      float dc = sp[4] - miou;
      l += dc * dc;                                  // contain_loss
      int lab = (int)r[1];
      #pragma unroll
      for (int c2 = 0; c2 < NCLS; ++c2) {            // class_loss
        float d = p[5*NBOX + c2] - ((c2 == lab) ? 1.0f : 0.0f);
        l += d * d;
      }
    } else {
      l = 0.5f * (p[4]*p[4] + p[9]*p[9]);            // L_NOOBJ * conf^2
    }
  }

  // --- reduce: WMMA wave sum -> LDS -> one partial per block (no atomics) ---
  float tot = wave_reduce32(l);                      // EXEC all-ones here
  if ((threadIdx.x & 31) == 0) wsum[threadIdx.x >> 5] = tot;
  __syncthreads();
  if (threadIdx.x == 0) {
    float s = 0.0f;
    #pragma unroll
    for (int w = 0; w < BLOCK / 32; ++w) s += wsum[w];
    partials[blockIdx.x] = s;
  }
}

// ---- kernel 4: deterministic final reduction (single block) -----------------
__global__ __launch_bounds__(BLOCK)
void yolo_finalize(const float* __restrict__ partials, int n,
                   float* __restrict__ out, float inv_batch) {
  __shared__ float wsum[BLOCK / 32];
  float s = 0.0f;
  for (int i = threadIdx.x; i < n; i += BLOCK) s += partials[i]; // fixed order
  float tot = wave_reduce32(s);                      // converged: EXEC all-ones
  if ((threadIdx.x & 31) == 0) wsum[threadIdx.x >> 5] = tot;
  __syncthreads();
  if (threadIdx.x == 0) {
    float t = 0.0f;
    #pragma unroll
    for (int w = 0; w < BLOCK / 32; ++w) t += wsum[w];
    out[0] = t * inv_batch;                          // (... ) / batch
  }
}

extern "C" void kernel_launch(void* const* d_in, const int* in_sizes, int n_in,
                              void* d_out, int out_size, void* d_ws, size_t ws_size,
                              hipStream_t stream) {
  const float* pred    = (const float*)d_in[0];
  const float* targets = (const float*)d_in[1];
  float*       out     = (float*)d_out;
  int ncells = in_sizes[0] / DCH;        // BATCH * 196
  int ntgt   = in_sizes[1] / 6;          // BATCH * 16
  int batch  = ncells / CPI;
  int nblk   = (ncells + BLOCK - 1) / BLOCK;

  int*   cellmap  = (int*)d_ws;                 // ncells ints (~800 KB)
  float* partials = (float*)((char*)d_ws + (size_t)ncells * sizeof(int));

  yolo_zero     <<<(ncells + 255) / 256, 256, 0, stream>>>(cellmap, ncells);
  yolo_encode   <<<(ntgt   + 255) / 256, 256, 0, stream>>>(targets, ntgt, cellmap, ncells);
  yolo_loss_main<<<nblk, BLOCK, 0, stream>>>(pred, targets, cellmap, partials, ncells);
  yolo_finalize <<<1, BLOCK, 0, stream>>>(partials, nblk, out, 1.0f / (float)batch);
}